// CachedMultiHeadedAttention_50431505990009
// MI455X (gfx1250) — compile-verified
//
#include <hip/hip_runtime.h>
#include <math.h>

typedef float v2f __attribute__((ext_vector_type(2)));
typedef float v8f __attribute__((ext_vector_type(8)));

#define NH      32
#define DMODEL  4096
#define DK      128
#define MAXSEQ  8192
#define SQ      1024
#define LMAX    4096

#define KC       32      // K-chunk staged in LDS for the big GEMMs
#define BSTRIDE  34      // padded LDS row stride (even => 8B-aligned b64 reads,
                         // 34*c mod 64 spreads banks across the 16 lanes)

__device__ __forceinline__ v8f wmma_f32(v2f a, v2f b, v8f c) {
  // V_WMMA_F32_16X16X4_F32 : D = A(16x4) * B(4x16) + C(16x16), fp32
  return __builtin_amdgcn_wmma_f32_16x16x4_f32(
      /*neg_a=*/false, a, /*neg_b=*/false, b,
      /*c_mod=*/(short)0, c, /*reuse_a=*/false, /*reuse_b=*/false);
}

// ---------------------------------------------------------------------------
// K1: q_scaled[h][s][k] = (x[s,:] @ Wq[h,:,:] + bq[h,:]) * (1/sqrt(DK))
// grid (SQ/256, DK/64, NH), block 256 (8 waves); wave computes 32x64.
// B is staged through LDS (transposed, padded) in KC-deep chunks.
// ---------------------------------------------------------------------------
__global__ void qproj_kernel(const float* __restrict__ x,
                             const float* __restrict__ Wq,
                             const float* __restrict__ bq,
                             float* __restrict__ q, float scale) {
  __shared__ float Bt[64 * BSTRIDE];
  const int tid  = threadIdx.x;
  const int wave = tid >> 5, lane = tid & 31;
  const int nl   = lane & 15, hf = lane >> 4;
  const int m0   = blockIdx.x * 256 + wave * 32;
  const int n0   = blockIdx.y * 64;
  const int h    = blockIdx.z;
  const float* B     = Wq + (size_t)h * DMODEL * DK;   // row-major [K][N=DK]
  const float* arow0 = x + (size_t)(m0 + nl) * DMODEL;
  const float* arow1 = x + (size_t)(m0 + 16 + nl) * DMODEL;

  v8f acc[2][4] = {};
  for (int k0 = 0; k0 < DMODEL; k0 += KC) {
    __syncthreads();
    // cooperative coalesced stage: 32x64 B tile -> LDS transposed [n][k]
#pragma unroll
    for (int r = 0; r < 2; ++r) {
      const int f  = (r * 256 + tid) * 4;
      const int kk = f >> 6, n = f & 63;
      const float4 vv = *(const float4*)(B + (size_t)(k0 + kk) * DK + n0 + n);
      Bt[(n + 0) * BSTRIDE + kk] = vv.x;
      Bt[(n + 1) * BSTRIDE + kk] = vv.y;
      Bt[(n + 2) * BSTRIDE + kk] = vv.z;
      Bt[(n + 3) * BSTRIDE + kk] = vv.w;
    }
    __builtin_prefetch(arow0 + k0 + KC, 0, 0);   // global_prefetch_b8
    __builtin_prefetch(arow1 + k0 + KC, 0, 0);
    __syncthreads();
#pragma unroll 4
    for (int kl = 0; kl < KC; kl += 4) {
      const int kk = kl + hf * 2;
      const v2f a0 = *(const v2f*)(arow0 + k0 + kk);
      const v2f a1 = *(const v2f*)(arow1 + k0 + kk);
#pragma unroll
      for (int j = 0; j < 4; ++j) {
        const v2f b = *(const v2f*)&Bt[(j * 16 + nl) * BSTRIDE + kk];
        acc[0][j] = wmma_f32(a0, b, acc[0][j]);
        acc[1][j] = wmma_f32(a1, b, acc[1][j]);
      }
    }
  }
#pragma unroll
  for (int mi = 0; mi < 2; ++mi)
#pragma unroll
    for (int j = 0; j < 4; ++j) {
      const int col  = n0 + j * 16 + nl;
      const float bb = bq[h * DK + col];
#pragma unroll
      for (int v = 0; v < 8; ++v) {
        const int mrow = m0 + mi * 16 + v + 8 * hf;
        q[((size_t)h * SQ + mrow) * DK + col] = (acc[mi][j][v] + bb) * scale;
      }
    }
}

// ---------------------------------------------------------------------------
// K2: k_new/v_new from the last token (GEMV). grid NH, block DK.
// ---------------------------------------------------------------------------
__global__ void kvnew_kernel(const float* __restrict__ x,
                             const float* __restrict__ Wk, const float* __restrict__ bk,
                             const float* __restrict__ Wv, const float* __restrict__ bv,
                             float* __restrict__ k_new, float* __restrict__ v_new) {
  const int h = blockIdx.x, t = threadIdx.x;
  const float* xl = x + (size_t)(SQ - 1) * DMODEL;
  const float* wk = Wk + (size_t)h * DMODEL * DK + t;
  const float* wv = Wv + (size_t)h * DMODEL * DK + t;
  float ak = 0.f, av = 0.f;
  for (int d = 0; d < DMODEL; ++d) {
    const float xv = xl[d];
    ak += xv * wk[(size_t)d * DK];
    av += xv * wv[(size_t)d * DK];
  }
  k_new[h * DK + t] = ak + bk[h * DK + t];
  v_new[h * DK + t] = av + bv[h * DK + t];
}

// ---------------------------------------------------------------------------
// K3: colsum[h][l] = sum_s exp(q_scaled[h,s,:] . keys[h,l,:])
// (reference softmax is over axis=2 == s). grid (LMAX/16, NH), block 256.
// K-row B-fragments are loop-invariant across the s-tiles handled by a wave:
// hoisted into 32 v2f registers and reused for all 8 s-tile iterations.
// ---------------------------------------------------------------------------
__global__ void colsum_kernel(const float* __restrict__ q,
                              const float* __restrict__ k_cache,
                              const float* __restrict__ k_new,
                              const int* __restrict__ pos_p,
                              float* __restrict__ colsum) {
  __shared__ float csum[16];
  const int posv = pos_p[0];
  const int L    = posv + 1;
  const int l0   = blockIdx.x * 16;
  const int h    = blockIdx.y;
  if (threadIdx.x < 16) csum[threadIdx.x] = 0.f;
  __syncthreads();

  if (l0 < L) {                               // block-uniform guard
    const int wave = threadIdx.x >> 5;
    const int lane = threadIdx.x & 31;
    const int nl   = lane & 15;
    const int hf   = lane >> 4;
    const int kk_off = hf * 2;
    const int l = l0 + nl;
    const float* krow = (l < posv)
        ? (k_cache + ((size_t)h * MAXSEQ + l) * DK)
        : (k_new + (size_t)h * DK);           // l >= L columns are never used

    v2f bfrag[DK / 4];                        // hoisted K fragments (64 VGPRs)
#pragma unroll
    for (int ks = 0; ks < DK / 4; ++ks)
      bfrag[ks] = *(const v2f*)(krow + ks * 4 + kk_off);

    for (int st = wave; st < SQ / 16; st += 8) {
      const float* qa = q + ((size_t)h * SQ + st * 16 + nl) * DK;
      v8f c = {};
#pragma unroll
      for (int ks = 0; ks < DK / 4; ++ks) {
        const v2f a = *(const v2f*)(qa + ks * 4 + kk_off);
        c = wmma_f32(a, bfrag[ks], c);
      }
      float p = 0.f;
#pragma unroll
      for (int v = 0; v < 8; ++v) p += __expf(c[v]);
      atomicAdd(&csum[nl], p);                // ds_add_f32
    }
  }
  __syncthreads();
  if (threadIdx.x < 16)
    colsum[(size_t)h * LMAX + l0 + threadIdx.x] = csum[threadIdx.x];
}

// ---------------------------------------------------------------------------
// K4: ctx[h][s][k] += sum_{l in split} (exp(score)/colsum[h,l]) * vals[h,l,k]
// l-range split 4 ways over blockIdx.z; partial contexts accumulate into the
// pre-zeroed concat buffer with global f32 atomics.
// The wave's q A-fragments are invariant across the whole l loop: hoisted
// into 32 v2f registers. Score tiles go C-layout -> A-layout via 1 KB LDS.
// grid (SQ/16, NH, 4), block 32 (1 wave).
// ---------------------------------------------------------------------------
#define LSPLIT 4
__global__ void ctx_kernel(const float* __restrict__ q,
                           const float* __restrict__ k_cache,
                           const float* __restrict__ v_cache,
                           const float* __restrict__ k_new,
                           const float* __restrict__ v_new,
                           const float* __restrict__ colsum,
                           const int* __restrict__ pos_p,
                           float* __restrict__ concat) {
  __shared__ float wtile[16 * 16];
  const int posv = pos_p[0];
  const int L    = posv + 1;
  const int nt   = (L + 15) >> 4;
  const int h    = blockIdx.y;
  const int m0   = blockIdx.x * 16;
  const int lt0  = blockIdx.z * (LMAX / 16 / LSPLIT);
  const int lt1  = min(nt, lt0 + (LMAX / 16 / LSPLIT));
  const int lane = threadIdx.x & 31;
  const int nl   = lane & 15;
  const int hf   = lane >> 4;
  const int kk_off = hf * 2;
  const float* qa = q + ((size_t)h * SQ + m0 + nl) * DK;

  if (lt0 >= lt1) return;                     // block-uniform

  v2f afrag[DK / 4];                          // hoisted q fragments (64 VGPRs)
#pragma unroll
  for (int ks = 0; ks < DK / 4; ++ks)
    afrag[ks] = *(const v2f*)(qa + ks * 4 + kk_off);

  v8f acc[8] = {};
  for (int lt = lt0; lt < lt1; ++lt) {
    const int l0 = lt * 16;
    const int l  = l0 + nl;
    // ---- scores tile (16s x 16l), K = DK via WMMA ----
    const float* krow = (l < posv)
        ? (k_cache + ((size_t)h * MAXSEQ + l) * DK)
        : (k_new + (size_t)h * DK);
    v8f c = {};
#pragma unroll
    for (int ks = 0; ks < DK / 4; ++ks) {
      const v2f b = *(const v2f*)(krow + ks * 4 + kk_off);
      c = wmma_f32(afrag[ks], b, c);
    }
    // ---- weights = exp(score)/colsum, masked for l >= L ----
    const float inv = (l < L) ? (1.0f / colsum[(size_t)h * LMAX + l]) : 0.0f;
    __syncthreads();                          // previous-iter LDS reads done
#pragma unroll
    for (int v = 0; v < 8; ++v)
      wtile[(v + 8 * hf) * 16 + nl] = __expf(c[v]) * inv;
    __syncthreads();
    // ---- ctx += W(16x16) @ V(16x128) via WMMA, K-dim = l ----
#pragma unroll
    for (int t = 0; t < 4; ++t) {
      const int kk = t * 4 + kk_off;          // l-sub index 0..15
      const v2f a = *(const v2f*)&wtile[nl * 16 + kk];
      const int lr0 = l0 + kk, lr1 = l0 + kk + 1;
      const float* vrow0 = (lr0 < posv)
          ? (v_cache + ((size_t)h * MAXSEQ + lr0) * DK) : (v_new + (size_t)h * DK);
      const float* vrow1 = (lr1 < posv)
          ? (v_cache + ((size_t)h * MAXSEQ + lr1) * DK) : (v_new + (size_t)h * DK);
#pragma unroll
      for (int j = 0; j < 8; ++j) {
        const int col = j * 16 + nl;
        v2f b; b.x = vrow0[col]; b.y = vrow1[col];
        acc[j] = wmma_f32(a, b, acc[j]);
      }
    }
  }
#pragma unroll
  for (int j = 0; j < 8; ++j)
#pragma unroll
    for (int v = 0; v < 8; ++v) {
      const int mrow = m0 + v + 8 * hf;
      atomicAdd(&concat[(size_t)mrow * DMODEL + h * DK + j * 16 + nl], acc[j][v]);
    }
}

// ---------------------------------------------------------------------------
// K5: out = concat @ Wo + bo  (1024x4096 @ 4096x4096)
// grid (SQ/256, DMODEL/64), block 256; wave computes 32x64; LDS-staged B.
// ---------------------------------------------------------------------------
__global__ void outproj_kernel(const float* __restrict__ concat,
                               const float* __restrict__ Wo,
                               const float* __restrict__ bo,
                               float* __restrict__ out) {
  __shared__ float Bt[64 * BSTRIDE];
  const int tid  = threadIdx.x;
  const int wave = tid >> 5, lane = tid & 31;
  const int nl   = lane & 15, hf = lane >> 4;
  const int m0   = blockIdx.x * 256 + wave * 32;
  const int n0   = blockIdx.y * 64;
  const float* arow0 = concat + (size_t)(m0 + nl) * DMODEL;
  const float* arow1 = concat + (size_t)(m0 + 16 + nl) * DMODEL;

  v8f acc[2][4] = {};
  for (int k0 = 0; k0 < DMODEL; k0 += KC) {
    __syncthreads();
#pragma unroll
    for (int r = 0; r < 2; ++r) {
      const int f  = (r * 256 + tid) * 4;
      const int kk = f >> 6, n = f & 63;
      const float4 vv = *(const float4*)(Wo + (size_t)(k0 + kk) * DMODEL + n0 + n);
      Bt[(n + 0) * BSTRIDE + kk] = vv.x;
      Bt[(n + 1) * BSTRIDE + kk] = vv.y;
      Bt[(n + 2) * BSTRIDE + kk] = vv.z;
      Bt[(n + 3) * BSTRIDE + kk] = vv.w;
    }
    __builtin_prefetch(arow0 + k0 + KC, 0, 0);
    __builtin_prefetch(arow1 + k0 + KC, 0, 0);
    __syncthreads();
#pragma unroll 4
    for (int kl = 0; kl < KC; kl += 4) {
      const int kk = kl + hf * 2;
      const v2f a0 = *(const v2f*)(arow0 + k0 + kk);
      const v2f a1 = *(const v2f*)(arow1 + k0 + kk);
#pragma unroll
      for (int j = 0; j < 4; ++j) {
        const v2f b = *(const v2f*)&Bt[(j * 16 + nl) * BSTRIDE + kk];
        acc[0][j] = wmma_f32(a0, b, acc[0][j]);
        acc[1][j] = wmma_f32(a1, b, acc[1][j]);
      }
    }
  }
#pragma unroll
  for (int mi = 0; mi < 2; ++mi)
#pragma unroll
    for (int j = 0; j < 4; ++j) {
      const int col  = n0 + j * 16 + nl;
      const float bb = bo[col];
#pragma unroll
      for (int v = 0; v < 8; ++v)
        out[(size_t)(m0 + mi * 16 + v + 8 * hf) * DMODEL + col] = acc[mi][j][v] + bb;
    }
}

// ---------------------------------------------------------------------------
extern "C" void kernel_launch(void* const* d_in, const int* in_sizes, int n_in,
                              void* d_out, int out_size, void* d_ws, size_t ws_size,
                              hipStream_t stream) {
  const float* x       = (const float*)d_in[0];
  const float* k_cache = (const float*)d_in[1];
  const float* v_cache = (const float*)d_in[2];
  const float* Wq      = (const float*)d_in[3];
  const float* bq      = (const float*)d_in[4];
  const float* Wk      = (const float*)d_in[5];
  const float* bk      = (const float*)d_in[6];
  const float* Wv      = (const float*)d_in[7];
  const float* bv      = (const float*)d_in[8];
  const float* Wo      = (const float*)d_in[9];
  const float* bo      = (const float*)d_in[10];
  const int*   pos     = (const int*)d_in[11];
  float* out = (float*)d_out;

  // workspace layout (floats)
  float* q      = (float*)d_ws;                    // NH*SQ*DK   = 4,194,304
  float* knew   = q      + (size_t)NH * SQ * DK;   // NH*DK      = 4096
  float* vnew   = knew   + NH * DK;                // NH*DK      = 4096
  float* colsum = vnew   + NH * DK;                // NH*LMAX    = 131,072
  float* concat = colsum + (size_t)NH * LMAX;      // SQ*DMODEL  = 4,194,304

  const float scale = 0.088388347648318447f;       // 1/sqrt(128)

  qproj_kernel  <<<dim3(SQ / 256, DK / 64, NH), 256, 0, stream>>>(x, Wq, bq, q, scale);
  kvnew_kernel  <<<dim3(NH), DK, 0, stream>>>(x, Wk, bk, Wv, bv, knew, vnew);
  colsum_kernel <<<dim3(LMAX / 16, NH), 256, 0, stream>>>(q, k_cache, knew, pos, colsum);
  hipMemsetAsync(concat, 0, (size_t)SQ * DMODEL * sizeof(float), stream);
  ctx_kernel    <<<dim3(SQ / 16, NH, LSPLIT), 32, 0, stream>>>(q, k_cache, v_cache,
                                                               knew, vnew, colsum, pos, concat);
  outproj_kernel<<<dim3(SQ / 256, DMODEL / 64), 256, 0, stream>>>(concat, Wo, bo, out);
}